// CharRNN_28398323761472
// MI455X (gfx1250) — compile-verified
//
#include <hip/hip_runtime.h>

// ---------------------------------------------------------------------------
// 3-layer shared-weight LSTM (B=50, T=8192, H=65) + dense projection.
// Persistent-workgroup design: 4 blocks x 544 threads (17 waves, wave32).
// Each block owns 16 batch rows and runs the full 8192-step recurrence.
//   - All weights staged once into LDS (~214 KB of the 320 KB WGP LDS),
//     then each wave HOISTS its loop-invariant WMMA B-fragments (W, U, Wd
//     columns) into VGPRs -- zero weight LDS traffic in the steady state.
//   - Each of the 17 waves owns exactly one 16-wide N tile of the 260-wide
//     gate matrix, so the z-GEMM is branch-free (EXEC all-1s for WMMA).
//   - V_WMMA_F32_16X16X4_F32 (full fp32 tensor path): the cell state
//     integrates over 8192 steps and the kernel is latency-bound, so the
//     high-precision WMMA variant is free relative to the dependency chain.
// ---------------------------------------------------------------------------

#define B_SZ   50
#define T_LEN  8192
#define H_DIM  65
#define G_DIM  260            // 4*H (gates i,f,g,o)
#define KP     68             // K padded to 17*4 (WMMA K=4 tiles)
#define NPZ    272            // gate cols padded to 17*16
#define NPD    80             // dense cols padded to 5*16
#define KT     17             // K tiles
#define NTZ    17             // N tiles for gates
#define NTD    5              // N tiles for dense proj
#define THREADS 544           // 17 waves: one per gate N-tile
#define NWAVE  17

typedef __attribute__((ext_vector_type(2))) float v2f;
typedef __attribute__((ext_vector_type(8))) float v8f;

// LDS layout (float offsets)
#define OFF_W   0
#define OFF_U   (OFF_W  + KP*NPZ)       // 18496
#define OFF_WD  (OFF_U  + KP*NPZ)
#define OFF_B   (OFF_WD + KP*NPD)
#define OFF_BD  (OFF_B  + NPZ)
#define OFF_H   (OFF_BD + NPD)          // 3 layers x 16 rows x KP
#define OFF_C   (OFF_H  + 3*16*KP)
#define OFF_X   (OFF_C  + 3*16*KP)
#define OFF_Z   (OFF_X  + 16*KP)
#define LDS_FLOATS (OFF_Z + 16*NPZ)     // 54752 floats = 219,008 bytes

__device__ __forceinline__ float sigmoid_(float v) {
    return 1.0f / (1.0f + __expf(-v));
}
__device__ __forceinline__ float tanh_(float v) {
    return 2.0f / (1.0f + __expf(-2.0f * v)) - 1.0f;
}

// One LSTM cell step for a 16-row batch tile.
// z(:, col-tile) = Xb@W + Hb@U + b via WMMA with register-resident B frags,
// then gate nonlinearities update Hb/Cb in place. Xb/Hb/Cb are [16][KP].
__device__ __forceinline__ void lstm_layer(float* smem, const float* Xb,
                                           float* Hb, float* Cb,
                                           const v2f* fW, const v2f* fU,
                                           int col, int mrow, int half) {
    float* sZ = smem + OFF_Z;
    const float* sB = smem + OFF_B;

    // Fold bias into the WMMA C operand (bias depends only on col = lane%16)
    const float bias = sB[col];
    v8f acc;
    #pragma unroll
    for (int j = 0; j < 8; ++j) acc[j] = bias;

    const float* Xrow = Xb + mrow * KP;
    const float* Hrow = Hb + mrow * KP;
    #pragma unroll
    for (int kt = 0; kt < KT; ++kt) {          // x @ W
        v2f a = *(const v2f*)(Xrow + kt * 4 + 2 * half);
        acc = __builtin_amdgcn_wmma_f32_16x16x4_f32(
            false, a, false, fW[kt], (short)0, acc, false, false);
    }
    #pragma unroll
    for (int kt = 0; kt < KT; ++kt) {          // h @ U
        v2f a = *(const v2f*)(Hrow + kt * 4 + 2 * half);
        acc = __builtin_amdgcn_wmma_f32_16x16x4_f32(
            false, a, false, fU[kt], (short)0, acc, false, false);
    }
    #pragma unroll
    for (int j = 0; j < 8; ++j)                // C layout: row = j + 8*half
        sZ[(j + 8 * half) * NPZ + col] = acc[j];
    __syncthreads();

    // Gate nonlinearities + state update (i,f,g,o order as in Keras LSTMCell)
    for (int idx = threadIdx.x; idx < 16 * H_DIM; idx += THREADS) {
        const int r = idx / H_DIM, u = idx % H_DIM;
        const float iv = sigmoid_(sZ[r * NPZ + u]);
        const float fv = sigmoid_(sZ[r * NPZ + u + H_DIM]);
        const float gv = tanh_   (sZ[r * NPZ + u + 2 * H_DIM]);
        const float ov = sigmoid_(sZ[r * NPZ + u + 3 * H_DIM]);
        const float cn = fv * Cb[r * KP + u] + iv * gv;
        Cb[r * KP + u] = cn;
        Hb[r * KP + u] = ov * tanh_(cn);
    }
    __syncthreads();
}

__global__ __launch_bounds__(THREADS)
void lstm3_shared_kernel(const float* __restrict__ x, const float* __restrict__ W,
                         const float* __restrict__ U, const float* __restrict__ b,
                         const float* __restrict__ Wd, const float* __restrict__ bd,
                         float* __restrict__ out) {
    extern __shared__ float smem[];
    const int tid = threadIdx.x;
    const int wave = tid >> 5, lane = tid & 31;
    const int half = lane >> 4, mrow = lane & 15;
    const int bbase = blockIdx.x * 16;
    const int col = wave * 16 + mrow;            // this wave's gate column

    // ---- Stage weights into LDS (zero-padded to WMMA tile multiples) ----
    for (int i = tid; i < KP * NPZ; i += THREADS) {
        const int k = i / NPZ, n = i % NPZ;
        const bool ok = (k < H_DIM) && (n < G_DIM);
        smem[OFF_W + i] = ok ? W[k * G_DIM + n] : 0.0f;
        smem[OFF_U + i] = ok ? U[k * G_DIM + n] : 0.0f;
    }
    for (int i = tid; i < KP * NPD; i += THREADS) {
        const int k = i / NPD, n = i % NPD;
        smem[OFF_WD + i] = (k < H_DIM && n < H_DIM) ? Wd[k * H_DIM + n] : 0.0f;
    }
    for (int i = tid; i < NPZ; i += THREADS)
        smem[OFF_B + i] = (i < G_DIM) ? b[i] : 0.0f;
    for (int i = tid; i < NPD; i += THREADS)
        smem[OFF_BD + i] = (i < H_DIM) ? bd[i] : 0.0f;
    for (int i = tid; i < 3 * 16 * KP; i += THREADS) {
        smem[OFF_H + i] = 0.0f;          // zero initial h (pads stay zero)
        smem[OFF_C + i] = 0.0f;          // zero initial c
    }
    for (int i = tid; i < 16 * KP; i += THREADS)
        smem[OFF_X + i] = 0.0f;          // xt pad columns stay zero
    __syncthreads();

    // ---- Hoist loop-invariant WMMA B-fragments (weights) into VGPRs ----
    // B (4x16 f32) fragment: lane%16 = N, element j holds K = kt*4 + 2*half + j
    v2f fW[KT], fU[KT], fD[KT];
    {
        const float* sW  = smem + OFF_W;
        const float* sU  = smem + OFF_U;
        const float* sWd = smem + OFF_WD;
        #pragma unroll
        for (int kt = 0; kt < KT; ++kt) {
            const int k = kt * 4 + 2 * half;
            fW[kt].x = sW[(k + 0) * NPZ + col];
            fW[kt].y = sW[(k + 1) * NPZ + col];
            fU[kt].x = sU[(k + 0) * NPZ + col];
            fU[kt].y = sU[(k + 1) * NPZ + col];
        }
        if (wave < NTD) {                         // dense-proj waves only
            #pragma unroll
            for (int kt = 0; kt < KT; ++kt) {
                const int k = kt * 4 + 2 * half;
                fD[kt].x = sWd[(k + 0) * NPD + col];
                fD[kt].y = sWd[(k + 1) * NPD + col];
            }
        }
    }
    __syncthreads();

    float* sX  = smem + OFF_X;
    float* sH0 = smem + OFF_H;
    float* sH1 = sH0 + 16 * KP;
    float* sH2 = sH1 + 16 * KP;
    float* sC0 = smem + OFF_C;
    float* sC1 = sC0 + 16 * KP;
    float* sC2 = sC1 + 16 * KP;
    const float* sBd = smem + OFF_BD;

    // ---- Sequential recurrence over T ----
    for (int t = 0; t < T_LEN; ++t) {
        // Stage x[:, t, :] for this batch tile (rows >= B zeroed)
        for (int i = tid; i < 16 * H_DIM; i += THREADS) {
            const int r = i / H_DIM, u = i % H_DIM;
            const int bb = bbase + r;
            sX[r * KP + u] =
                (bb < B_SZ) ? x[((size_t)bb * T_LEN + t) * H_DIM + u] : 0.0f;
        }
        // Prefetch next timestep's rows (global_prefetch_b8)
        if (t + 1 < T_LEN && lane == 0 && wave < 16) {
            const int bb = bbase + wave;
            if (bb < B_SZ)
                __builtin_prefetch(&x[((size_t)bb * T_LEN + (t + 1)) * H_DIM], 0, 1);
        }
        __syncthreads();

        // Three stacked layers, identical (shared) weights
        lstm_layer(smem, sX,  sH0, sC0, fW, fU, col, mrow, half);
        lstm_layer(smem, sH0, sH1, sC1, fW, fU, col, mrow, half);
        lstm_layer(smem, sH1, sH2, sC2, fW, fU, col, mrow, half);

        // Dense projection: y = h2 @ Wd + bd, streamed straight to HBM
        if (wave < NTD) {                           // wave-uniform branch
            v8f acc;
            const float bias = sBd[col];
            #pragma unroll
            for (int j = 0; j < 8; ++j) acc[j] = bias;
            const float* Hrow = sH2 + mrow * KP;
            #pragma unroll
            for (int kt = 0; kt < KT; ++kt) {
                v2f a = *(const v2f*)(Hrow + kt * 4 + 2 * half);
                acc = __builtin_amdgcn_wmma_f32_16x16x4_f32(
                    false, a, false, fD[kt], (short)0, acc, false, false);
            }
            if (col < H_DIM) {
                #pragma unroll
                for (int j = 0; j < 8; ++j) {
                    const int row = j + 8 * half;
                    const int bb = bbase + row;
                    if (bb < B_SZ)
                        out[((size_t)bb * T_LEN + t) * H_DIM + col] = acc[j];
                }
            }
        }
        __syncthreads();
    }
}

extern "C" void kernel_launch(void* const* d_in, const int* in_sizes, int n_in,
                              void* d_out, int out_size, void* d_ws, size_t ws_size,
                              hipStream_t stream) {
    (void)in_sizes; (void)n_in; (void)d_ws; (void)ws_size; (void)out_size;
    const float* x  = (const float*)d_in[0];
    const float* W  = (const float*)d_in[1];
    const float* U  = (const float*)d_in[2];
    const float* b  = (const float*)d_in[3];
    const float* Wd = (const float*)d_in[4];
    const float* bd = (const float*)d_in[5];
    float* out = (float*)d_out;

    const dim3 grid((B_SZ + 15) / 16);              // 4 persistent workgroups
    const size_t shmem = (size_t)LDS_FLOATS * sizeof(float);  // ~214 KB (CDNA5 WGP LDS)
    lstm3_shared_kernel<<<grid, dim3(THREADS), shmem, stream>>>(x, W, U, b, Wd, bd, out);
}